// Model_54915451846792
// MI455X (gfx1250) — compile-verified
//
#include <hip/hip_runtime.h>
#include <math.h>

// ---------------- model constants ----------------
#define N_ATOMS   128
#define NSAMP     32768
#define NFFT      65536          // 256^2 -> two fused radix-16 (radix-256) passes
#define TOTAL_RES 4096
#define N_ROOMS   8
#define INV_NFFT  (1.0f / 65536.0f)
#define PI_F      3.14159265358979323846f
#define INV_SQRT_2PI 0.39894228040143267794f

typedef float v2f __attribute__((ext_vector_type(2)));
typedef float v8f __attribute__((ext_vector_type(8)));

// SMALL param table: field f, atom a -> SM[f*128 + a]
// 0 mix0, 1 mix1, 2 wet0, 3 wet1, 4 amp, 5 logd_decay, 6 logd_filt,
// 7 nf_mean, 8 nf_std, 9 rf1_std, 10 rf2_std, 11 env_mean, 12 env_std, 13 norm
#define SMF 16

__device__ __forceinline__ float gauss_pdf(float f, float mean, float std) {
  float z = (f - mean) / std;
  return __expf(-0.5f * z * z) * INV_SQRT_2PI / std;
}

// exponential_decay: per-atom constant decay d => frame f value d^(f+1),
// linearly interpolated (align_corners=False) from 128 frames to 32768 samples.
__device__ __forceinline__ float decay_env_val(float logd, int t) {
  float pos = ((float)t + 0.5f) * (128.0f / 32768.0f) - 0.5f;
  pos = fminf(fmaxf(pos, 0.0f), 127.0f);
  float fl = floorf(pos);
  int lo = (int)fl;
  int hi = lo + 1; if (hi > 127) hi = 127;
  float w = pos - fl;
  return __expf(logd * (float)(lo + 1)) * (1.0f - w) +
         __expf(logd * (float)(hi + 1)) * w;
}

// 16x16 complex DFT on a 16-column batch via 16x V_WMMA_F32_16X16X4_F32.
// A = DFT-16 cos / sign*sin matrices (4 K-chunks), B = data (Re/Im).
__device__ __forceinline__ void dft16_wmma(const v2f* Acr, const v2f* Asi,
                                           const v2f* Br, const v2f* Bi,
                                           v8f& Dr, v8f& Di) {
  Dr = (v8f){}; Di = (v8f){};
  #pragma unroll
  for (int c = 0; c < 4; ++c)
    Dr = __builtin_amdgcn_wmma_f32_16x16x4_f32(false, Acr[c], false, Br[c],
                                               (short)0, Dr, false, false);
  #pragma unroll
  for (int c = 0; c < 4; ++c) {
    v2f An = -Asi[c];
    Dr = __builtin_amdgcn_wmma_f32_16x16x4_f32(false, An, false, Bi[c],
                                               (short)0, Dr, false, false);
  }
  #pragma unroll
  for (int c = 0; c < 4; ++c)
    Di = __builtin_amdgcn_wmma_f32_16x16x4_f32(false, Acr[c], false, Bi[c],
                                               (short)0, Di, false, false);
  #pragma unroll
  for (int c = 0; c < 4; ++c)
    Di = __builtin_amdgcn_wmma_f32_16x16x4_f32(false, Asi[c], false, Br[c],
                                               (short)0, Di, false, false);
}

// ---------------- fused radix-256 Stockham pass (two radix-16 stages) -------
// PASS 0: stages Ns=1 then Ns=16.  PASS 1: stages Ns=256 then Ns=4096.
// Block owns a closed 4096-point working set staged through padded LDS (34 KB).
// grid = (16, batch), block = 256 (8 waves); each wave: 2 tiles x 2 phases
// x 16 WMMAs. sign = -1 forward, +1 inverse.
#define LDSPAD(l) ((l) + ((l) >> 4))

template <int PASS>
__global__ __launch_bounds__(256)
void fft_fused256(const float2* __restrict__ in, float2* __restrict__ out,
                  float sign) {
  __shared__ float2 lds[4352];
  const int lane = threadIdx.x & 31;
  const int wave = threadIdx.x >> 5;
  const int blk  = blockIdx.x;            // 0..15
  const int b  = lane & 15;               // WMMA column handled by this lane
  const int hi = lane >> 4;
  const float2* __restrict__ src = in  + (size_t)blockIdx.y * NFFT;
  float2* __restrict__ dst       = out + (size_t)blockIdx.y * NFFT;

  // constant DFT-16 factors for this lane (shared by both phases)
  v2f Acr[4], Asi[4];
  #pragma unroll
  for (int c = 0; c < 4; ++c) {
    #pragma unroll
    for (int e = 0; e < 2; ++e) {
      int k = c * 4 + hi * 2 + e;
      float ta = (2.0f * PI_F / 16.0f) * (float)((b * k) & 15);
      float sn, cs;
      __sincosf(ta, &sn, &cs);
      Acr[c][e] = cs;
      Asi[c][e] = sign * sn;
    }
  }

  // ---- phase 1: first radix-16 stage, global -> LDS
  #pragma unroll
  for (int tt = 0; tt < 2; ++tt) {
    const int tau = wave + 8 * tt;        // 0..15
    const float ang = (PASS == 0)
        ? 0.0f
        : sign * 2.0f * PI_F * (float)(16 * blk + b) * (1.0f / 4096.0f);
    v2f Br[4], Bi[4];
    #pragma unroll
    for (int c = 0; c < 4; ++c) {
      #pragma unroll
      for (int e = 0; e < 2; ++e) {
        int k = c * 4 + hi * 2 + e;
        float2 x = src[4096 * k + 256 * tau + 16 * blk + b];
        float sn, cs;
        __sincosf(ang * (float)k, &sn, &cs);
        Br[c][e] = x.x * cs - x.y * sn;
        Bi[c][e] = x.x * sn + x.y * cs;
      }
    }
    v8f Dr, Di;
    dft16_wmma(Acr, Asi, Br, Bi, Dr, Di);
    #pragma unroll
    for (int p = 0; p < 8; ++p) {
      int r = p + 8 * hi;
      int l = (PASS == 0) ? (256 * tau + 16 * b + r)   // slot of 16*j1 + r
                          : (256 * tau + 16 * r + b);  // slot of stage-3 output
      lds[LDSPAD(l)] = make_float2(Dr[p], Di[p]);
    }
  }
  __syncthreads();

  // ---- phase 2: second radix-16 stage, LDS -> global
  #pragma unroll
  for (int tt = 0; tt < 2; ++tt) {
    const int tau = wave + 8 * tt;
    const float ang = (PASS == 0)
        ? sign * 2.0f * PI_F * (float)b * (1.0f / 256.0f)
        : sign * 2.0f * PI_F * (float)(256 * tau + 16 * blk + b) *
              (1.0f / 65536.0f);
    v2f Br[4], Bi[4];
    #pragma unroll
    for (int c = 0; c < 4; ++c) {
      #pragma unroll
      for (int e = 0; e < 2; ++e) {
        int k = c * 4 + hi * 2 + e;
        int l = 256 * k + 16 * tau + b;
        float2 x = lds[LDSPAD(l)];
        float sn, cs;
        __sincosf(ang * (float)k, &sn, &cs);
        Br[c][e] = x.x * cs - x.y * sn;
        Bi[c][e] = x.x * sn + x.y * cs;
      }
    }
    v8f Dr, Di;
    dft16_wmma(Acr, Asi, Br, Bi, Dr, Di);
    #pragma unroll
    for (int p = 0; p < 8; ++p) {
      int r = p + 8 * hi;
      int g = (PASS == 0) ? (4096 * blk + 256 * tau + 16 * r + b)
                          : (4096 * r + 256 * tau + 16 * blk + b);
      dst[g] = make_float2(Dr[p], Di[p]);
    }
  }
}

// ---------------- elementwise / reduction kernels ----------------

// pack real rows (len NSAMP) into zero-padded complex rows (len NFFT);
// rowidx==nullptr -> identity row mapping (gather path for one-hot einsums).
__global__ void pack_rows(const float* __restrict__ src,
                          const int* __restrict__ rowidx,
                          float2* __restrict__ dst) {
  int a = blockIdx.y;
  int t = blockIdx.x * blockDim.x + threadIdx.x;
  int row = rowidx ? rowidx[a] : a;
  float v = (t < NSAMP) ? src[(size_t)row * NSAMP + t] : 0.0f;
  dst[(size_t)a * NFFT + t] = make_float2(v, 0.0f);
}

// multiply spectrum by real symmetric gaussian filter over normalized freq.
__global__ void mul_gauss(const float2* __restrict__ in, float2* __restrict__ out,
                          const float* __restrict__ means,
                          const float* __restrict__ stds) {
  int a = blockIdx.y;
  int k = blockIdx.x * blockDim.x + threadIdx.x;
  int kk = (k <= NFFT / 2) ? k : (NFFT - k);
  float f = (float)kk / (float)(NFFT / 2);
  float mean = means ? means[a] : 0.0f;
  float h = gauss_pdf(f, mean, stds[a]);
  size_t i = (size_t)a * NFFT + k;
  float2 v = in[i];
  out[i] = make_float2(v.x * h, v.y * h);
}

// positioned noise: envelope * filtered noise, then one-hot shift by s.
__global__ void make_pn(const float2* __restrict__ fn, float* __restrict__ pn,
                        const int* __restrict__ shift_idx,
                        const float* __restrict__ SM) {
  int a = blockIdx.y;
  int t = blockIdx.x * blockDim.x + threadIdx.x;
  int s = shift_idx[a];
  float v = 0.0f;
  if (t >= s) {
    int u = t - s;
    float env = gauss_pdf((float)u / (float)(NSAMP - 1),
                          SM[11 * 128 + a], SM[12 * 128 + a]);
    v = fn[(size_t)a * NFFT + u].x * INV_NFFT * env;
  }
  pn[(size_t)a * NSAMP + t] = v;
}

// (inverse-FFT result) * decay_env -> repacked zero-padded complex for next FFT.
__global__ void decay_pack(const float2* __restrict__ in, float2* __restrict__ out,
                           const float* __restrict__ SM) {
  int a = blockIdx.y;
  int t = blockIdx.x * blockDim.x + threadIdx.x;
  float v = 0.0f;
  if (t < NSAMP) {
    float logd = SM[5 * 128 + a];
    v = in[(size_t)a * NFFT + t].x * INV_NFFT * decay_env_val(logd, t);
  }
  out[(size_t)a * NFFT + t] = make_float2(v, 0.0f);
}

// complex spectral multiply: xy *= z  (convolution theorem)
__global__ void cmul(float2* __restrict__ xy, const float2* __restrict__ z) {
  size_t i = (size_t)blockIdx.y * NFFT + blockIdx.x * blockDim.x + threadIdx.x;
  float2 x = xy[i], w = z[i];
  xy[i] = make_float2(x.x * w.x - x.y * w.y, x.x * w.y + x.y * w.x);
}

// mixed (+)= Re(ifft(conv)) * {xfade | 1-xfade}
__global__ void mix_accum(const float2* __restrict__ in, float* __restrict__ mixed,
                          const float* __restrict__ SM, int first, int oneMinus) {
  int a = blockIdx.y;
  int t = blockIdx.x * blockDim.x + threadIdx.x;
  float xf = decay_env_val(SM[6 * 128 + a], t);
  float w = oneMinus ? (1.0f - xf) : xf;
  float v = in[(size_t)a * NFFT + t].x * INV_NFFT * w;
  size_t o = (size_t)a * NSAMP + t;
  mixed[o] = first ? v : (mixed[o] + v);
}

// final = pn*mix0 + mixed*mix1 (stored in-place in mixed)
__global__ void final_pre(const float* __restrict__ pn, float* __restrict__ mixed,
                          const float* __restrict__ SM) {
  int a = blockIdx.y;
  int t = blockIdx.x * blockDim.x + threadIdx.x;
  size_t o = (size_t)a * NSAMP + t;
  mixed[o] = pn[o] * SM[0 * 128 + a] + mixed[o] * SM[1 * 128 + a];
}

__global__ void norm_kernel(const float* __restrict__ mixed, float* __restrict__ SM) {
  __shared__ float red[256];
  int a = blockIdx.x;
  float s = 0.0f;
  for (int t = threadIdx.x; t < NSAMP; t += blockDim.x) {
    float v = mixed[(size_t)a * NSAMP + t];
    s += v * v;
  }
  red[threadIdx.x] = s;
  __syncthreads();
  for (int st = 128; st > 0; st >>= 1) {
    if (threadIdx.x < st) red[threadIdx.x] += red[threadIdx.x + st];
    __syncthreads();
  }
  if (threadIdx.x == 0) SM[13 * 128 + a] = sqrtf(red[0]) + 1e-8f;
}

// unit-norm * |amp|, store back and pack complex for reverb FFT.
__global__ void scale_pack(float* __restrict__ mixed, float2* __restrict__ c,
                           const float* __restrict__ SM) {
  int a = blockIdx.y;
  int t = blockIdx.x * blockDim.x + threadIdx.x;
  float v = 0.0f;
  if (t < NSAMP) {
    size_t o = (size_t)a * NSAMP + t;
    v = mixed[o] / SM[13 * 128 + a] * SM[4 * 128 + a];
    mixed[o] = v;
  }
  c[(size_t)a * NFFT + t] = make_float2(v, 0.0f);
}

// spectrum *= precomputed room spectrum (hard one-hot room choice == gather)
__global__ void wet_mul(float2* __restrict__ c, const float2* __restrict__ rooms,
                        const int* __restrict__ room_idx) {
  int a = blockIdx.y;
  int k = blockIdx.x * blockDim.x + threadIdx.x;
  const float2* R = rooms + (size_t)room_idx[a] * NFFT;
  size_t i = (size_t)a * NFFT + k;
  float2 x = c[i], w = R[k];
  c[i] = make_float2(x.x * w.x - x.y * w.y, x.x * w.y + x.y * w.x);
}

__global__ void final_out(const float* __restrict__ mixed, const float2* __restrict__ c,
                          const float* __restrict__ SM, float* __restrict__ out) {
  int a = blockIdx.y;
  int t = blockIdx.x * blockDim.x + threadIdx.x;
  size_t o = (size_t)a * NSAMP + t;
  out[o] = mixed[o] * SM[2 * 128 + a] +
           c[(size_t)a * NFFT + t].x * INV_NFFT * SM[3 * 128 + a];
  if (t == 0) out[(size_t)N_ATOMS * NSAMP + a] = SM[4 * 128 + a];  // amps output
}

__global__ void argmax_kernel(const float* __restrict__ src, int len,
                              int* __restrict__ idxout) {
  __shared__ float sv[256];
  __shared__ int si[256];
  int a = blockIdx.x;
  const float* p = src + (size_t)a * len;
  float best = -INFINITY; int bi = 0;
  for (int t = threadIdx.x; t < len; t += blockDim.x) {
    float v = p[t];
    if (v > best) { best = v; bi = t; }
  }
  sv[threadIdx.x] = best; si[threadIdx.x] = bi;
  __syncthreads();
  for (int st = 128; st > 0; st >>= 1) {
    if (threadIdx.x < st) {
      bool take = (sv[threadIdx.x + st] > sv[threadIdx.x]) ||
                  (sv[threadIdx.x + st] == sv[threadIdx.x] &&
                   si[threadIdx.x + st] < si[threadIdx.x]);
      if (take) { sv[threadIdx.x] = sv[threadIdx.x + st]; si[threadIdx.x] = si[threadIdx.x + st]; }
    }
    __syncthreads();
  }
  if (threadIdx.x == 0) idxout[a] = si[0];
}

// per-atom scalar params + 4-wide LayerNorm/MLP reverb head
__global__ void small_kernel(const float* __restrict__ env, const float* __restrict__ mix,
                             const float* __restrict__ decays, const float* __restrict__ fdecays,
                             const float* __restrict__ nfilt, const float* __restrict__ rfilt,
                             const float* __restrict__ rfilt2, const float* __restrict__ amps,
                             const float* __restrict__ verb, const float* __restrict__ gamma,
                             const float* __restrict__ beta, const float* __restrict__ w1,
                             const float* __restrict__ b1, const float* __restrict__ w_room,
                             const float* __restrict__ b_room, const float* __restrict__ w_mix,
                             const float* __restrict__ b_mix,
                             float* __restrict__ SM, int* __restrict__ room_idx) {
  int a = blockIdx.x * blockDim.x + threadIdx.x;
  if (a >= N_ATOMS) return;
  // overall mix softmax
  float m0 = mix[a * 2], m1 = mix[a * 2 + 1];
  float mx = fmaxf(m0, m1);
  float e0 = __expf(m0 - mx), e1 = __expf(m1 - mx), es = e0 + e1;
  SM[0 * 128 + a] = e0 / es; SM[1 * 128 + a] = e1 / es;
  // LayerNorm(4) -> hidden -> room argmax + wet softmax
  float v[4]; float mu = 0.0f;
  for (int i = 0; i < 4; ++i) { v[i] = verb[a * 4 + i]; mu += v[i]; }
  mu *= 0.25f;
  float var = 0.0f;
  for (int i = 0; i < 4; ++i) { float d = v[i] - mu; var += d * d; }
  var *= 0.25f;
  float inv = rsqrtf(var + 1e-5f);
  float ctx[4], h[4];
  for (int i = 0; i < 4; ++i) ctx[i] = (v[i] - mu) * inv * gamma[i] + beta[i];
  for (int j = 0; j < 4; ++j) {
    float s = b1[j];
    for (int i = 0; i < 4; ++i) s += ctx[i] * w1[i * 4 + j];
    h[j] = (s > 0.0f) ? s : 0.01f * s;  // leaky_relu
  }
  int bi = 0; float bv = -INFINITY;
  for (int r = 0; r < N_ROOMS; ++r) {
    float s = b_room[r];
    for (int i = 0; i < 4; ++i) s += h[i] * w_room[i * N_ROOMS + r];
    if (s > bv) { bv = s; bi = r; }
  }
  room_idx[a] = bi;
  float l0 = b_mix[0], l1 = b_mix[1];
  for (int i = 0; i < 4; ++i) { l0 += h[i] * w_mix[i * 2]; l1 += h[i] * w_mix[i * 2 + 1]; }
  float wmx = fmaxf(l0, l1);
  float f0 = __expf(l0 - wmx), f1 = __expf(l1 - wmx), fs = f0 + f1;
  SM[2 * 128 + a] = f0 / fs; SM[3 * 128 + a] = f1 / fs;
  SM[4 * 128 + a] = fabsf(amps[a]);
  // decays: d = BASE + sigmoid(x)*(1-BASE)*0.95; cumsum(log) => (f+1)*log d
  float sg = 1.0f / (1.0f + __expf(-decays[a]));
  SM[5 * 128 + a] = logf(0.02f + sg * 0.931f + 1e-12f);
  sg = 1.0f / (1.0f + __expf(-fdecays[a]));
  SM[6 * 128 + a] = logf(0.02f + sg * 0.931f + 1e-12f);
  SM[7 * 128 + a] = nfilt[a * 2];
  SM[8 * 128 + a] = fabsf(nfilt[a * 2 + 1]) + 1e-12f;
  SM[9 * 128 + a] = fabsf(rfilt[a * 2 + 1]) + 1e-12f;
  SM[10 * 128 + a] = fabsf(rfilt2[a * 2 + 1]) + 1e-12f;
  SM[11 * 128 + a] = env[a * 2];
  SM[12 * 128 + a] = fabsf(env[a * 2 + 1] + 1e-12f) * 0.1f;
}

// ---------------- host orchestration ----------------
extern "C" void kernel_launch(void* const* d_in, const int* in_sizes, int n_in,
                              void* d_out, int out_size, void* d_ws, size_t ws_size,
                              hipStream_t stream) {
  (void)in_sizes; (void)n_in; (void)out_size; (void)ws_size;
  const float* in_env   = (const float*)d_in[1];
  const float* in_shift = (const float*)d_in[2];
  const float* in_mix   = (const float*)d_in[3];
  const float* in_dec   = (const float*)d_in[4];
  const float* in_fdec  = (const float*)d_in[5];
  const float* in_rch   = (const float*)d_in[6];
  const float* in_nf    = (const float*)d_in[7];
  const float* in_rf    = (const float*)d_in[8];
  const float* in_rf2   = (const float*)d_in[9];
  const float* in_amp   = (const float*)d_in[10];
  const float* in_verb  = (const float*)d_in[11];
  const float* in_gam   = (const float*)d_in[12];
  const float* in_bet   = (const float*)d_in[13];
  const float* in_w1    = (const float*)d_in[14];
  const float* in_b1    = (const float*)d_in[15];
  const float* in_wr    = (const float*)d_in[16];
  const float* in_br    = (const float*)d_in[17];
  const float* in_wm    = (const float*)d_in[18];
  const float* in_bm    = (const float*)d_in[19];
  const float* in_waves = (const float*)d_in[20];
  const float* in_noise = (const float*)d_in[21];
  const float* in_rooms = (const float*)d_in[22];
  float* out = (float*)d_out;

  // workspace carve-up (~229 MB; FFT ping-pong buffers stay L2-resident)
  char* base = (char*)d_ws;
  float2* c0 = (float2*)(base);                                 // 64 MB
  float2* c1 = (float2*)(base + ((size_t)64 << 20));            // 64 MB
  float2* c2 = (float2*)(base + ((size_t)128 << 20));           // 64 MB (F_pn)
  float2* cR = (float2*)(base + ((size_t)192 << 20));           // 4 MB room spectra
  float*  PN = (float*)(base + ((size_t)196 << 20));            // 16 MB
  float*  MIX = (float*)(base + ((size_t)212 << 20));           // 16 MB
  float*  SM = (float*)(base + ((size_t)228 << 20));            // SMF*128 floats
  int* res_idx   = (int*)(base + ((size_t)228 << 20) + SMF * 128 * sizeof(float));
  int* shift_idx = res_idx + 128;
  int* room_idx  = shift_idx + 128;

  dim3 b256(256);
  dim3 gF(NFFT / 256, N_ATOMS), gS(NSAMP / 256, N_ATOMS), gR(NFFT / 256, N_ROOMS);

  // full 65536-pt complex FFT = two fused radix-256 passes (4 radix-16 stages)
  auto fft = [&](float2* a, float2* b, float2* fin, float sign, int batch) {
    fft_fused256<0><<<dim3(16, batch), b256, 0, stream>>>(a, b, sign);
    fft_fused256<1><<<dim3(16, batch), b256, 0, stream>>>(b, fin, sign);
  };

  // scalar/per-atom params (hard one-hots -> index gathers)
  argmax_kernel<<<128, 256, 0, stream>>>(in_rch, TOTAL_RES, res_idx);
  argmax_kernel<<<128, 256, 0, stream>>>(in_shift, NSAMP, shift_idx);
  small_kernel<<<1, 128, 0, stream>>>(in_env, in_mix, in_dec, in_fdec, in_nf, in_rf,
                                      in_rf2, in_amp, in_verb, in_gam, in_bet, in_w1,
                                      in_b1, in_wr, in_br, in_wm, in_bm, SM, room_idx);

  // room IR spectra (8 unique rooms, computed once)
  pack_rows<<<gR, b256, 0, stream>>>(in_rooms, nullptr, cR);
  fft(cR, c0, cR, -1.0f, N_ROOMS);

  // noise path: FFT -> gaussian filter -> IFFT -> envelope+shift -> F_pn
  pack_rows<<<gF, b256, 0, stream>>>(in_noise, nullptr, c0);
  fft(c0, c1, c0, -1.0f, N_ATOMS);
  mul_gauss<<<gF, b256, 0, stream>>>(c0, c0, SM + 7 * 128, SM + 8 * 128);
  fft(c0, c1, c0, +1.0f, N_ATOMS);
  make_pn<<<gS, b256, 0, stream>>>(c0, PN, shift_idx, SM);
  pack_rows<<<gF, b256, 0, stream>>>(PN, nullptr, c0);
  fft(c0, c1, c2, -1.0f, N_ATOMS);                        // F_pn in c2

  // resonance path 1: gather wave row -> filter -> decay -> conv with pn
  pack_rows<<<gF, b256, 0, stream>>>(in_waves, res_idx, c0);
  fft(c0, c1, c0, -1.0f, N_ATOMS);
  mul_gauss<<<gF, b256, 0, stream>>>(c0, c0, nullptr, SM + 9 * 128);
  fft(c0, c1, c0, +1.0f, N_ATOMS);
  decay_pack<<<gF, b256, 0, stream>>>(c0, c0, SM);
  fft(c0, c1, c0, -1.0f, N_ATOMS);
  cmul<<<gF, b256, 0, stream>>>(c0, c2);
  fft(c0, c1, c0, +1.0f, N_ATOMS);
  mix_accum<<<gS, b256, 0, stream>>>(c0, MIX, SM, 1, 0);

  // resonance path 2 (second filter), weighted by 1 - xfade
  pack_rows<<<gF, b256, 0, stream>>>(in_waves, res_idx, c0);
  fft(c0, c1, c0, -1.0f, N_ATOMS);
  mul_gauss<<<gF, b256, 0, stream>>>(c0, c0, nullptr, SM + 10 * 128);
  fft(c0, c1, c0, +1.0f, N_ATOMS);
  decay_pack<<<gF, b256, 0, stream>>>(c0, c0, SM);
  fft(c0, c1, c0, -1.0f, N_ATOMS);
  cmul<<<gF, b256, 0, stream>>>(c0, c2);
  fft(c0, c1, c0, +1.0f, N_ATOMS);
  mix_accum<<<gS, b256, 0, stream>>>(c0, MIX, SM, 0, 1);

  // combine, normalize, apply amplitude, reverb, dry/wet mix
  final_pre<<<gS, b256, 0, stream>>>(PN, MIX, SM);
  norm_kernel<<<128, 256, 0, stream>>>(MIX, SM);
  scale_pack<<<gF, b256, 0, stream>>>(MIX, c0, SM);
  fft(c0, c1, c0, -1.0f, N_ATOMS);
  wet_mul<<<gF, b256, 0, stream>>>(c0, cR, room_idx);
  fft(c0, c1, c0, +1.0f, N_ATOMS);
  final_out<<<gS, b256, 0, stream>>>(MIX, c0, SM, out);
}